// BatchASTEncoder_13280038879631
// MI455X (gfx1250) — compile-verified
//
#include <hip/hip_runtime.h>

// ---------------------------------------------------------------------------
// BatchASTEncoder for MI455X (gfx1250, wave32).
// h = gather(encodes, node_ids) @ W_c^T + b_c          (P*B x E GEMM)
// 9 bottom-up tree levels: h[p] += (h[2p+1]+h[2p+2]) @ W_s^T + 2*b_s
// out = [relu(h)[postorder], max_p relu(h)]
// fp32 WMMA (V_WMMA_F32_16X16X4_F32) to stay faithful to the fp32 reference;
// global->LDS staging uses gfx1250 async-to-LDS copies, double-buffered.
// ---------------------------------------------------------------------------

#define TREE_DEPTH 10
#define P_NODES    1023           // 2^10 - 1
#define BATCH      64
#define E_DIM      256

typedef float v2f __attribute__((ext_vector_type(2)));
typedef float v8f __attribute__((ext_vector_type(8)));

#define KC    32                  // K chunk staged in LDS per pipeline stage
#define PITCH 36                  // LDS row pitch (floats): even (8B/16B aligned),
                                  // lane stride 36 %64 -> 16 distinct banks
#define NCHUNK (E_DIM / KC)       // 8

// ---- gfx1250 async global->LDS copy (ASYNCcnt-tracked, no VGPR round-trip) --
__device__ __forceinline__ void async_copy16(const void* gsrc, void* lds_dst) {
  // local->flat addrspacecast puts the LDS byte offset in the low 32 bits.
  unsigned lds_off = (unsigned)(unsigned long long)lds_dst;
  asm volatile("global_load_async_to_lds_b128 %0, %1, off"
               :: "v"(lds_off), "v"(gsrc)
               : "memory");
}
__device__ __forceinline__ void wait_async0() {
  asm volatile("s_wait_asynccnt 0" ::: "memory");
}

// GEMM core on staged LDS tiles.
//  Xs rows = M (batch rows), cols = K chunk
//  Ws rows = N (output cols), cols = K chunk  (B tile = W^T: B[k,n] = W[n,k])
// A frag (16x4 f32): lanes 0-15 M=lo K={kk,kk+1}; lanes 16-31 M=lo K={kk+2,kk+3}
// B frag (4x16 f32): lanes 0-15 N=lo K={kk,kk+1}; lanes 16-31 N=lo K={kk+2,kk+3}
__device__ __forceinline__ void gemm_chunk(const float (&Xs)[64][PITCH],
                                           const float (&Ws)[64][PITCH],
                                           int wave_m, int wave_n, int lane,
                                           v8f acc[2]) {
  const int hi = lane >> 4;
  const int lo = lane & 15;
#pragma unroll
  for (int kk = 0; kk < KC; kk += 4) {
    const int kcol = kk + 2 * hi;
    v2f a = *(const v2f*)&Xs[wave_m * 16 + lo][kcol];
#pragma unroll
    for (int t = 0; t < 2; ++t) {
      v2f b = *(const v2f*)&Ws[wave_n * 32 + t * 16 + lo][kcol];
      acc[t] = __builtin_amdgcn_wmma_f32_16x16x4_f32(
          /*neg_a=*/false, a, /*neg_b=*/false, b,
          /*c_mod=*/(short)0, acc[t], /*reuse_a=*/false, /*reuse_b=*/false);
    }
  }
}

// ---------------------------------------------------------------------------
// Phase 1: h[p,b,:] = encodes[node_ids[p,b],:] @ W_c^T + b_c
// grid = (P_NODES, E_DIM/64), block = 256 (8 waves: 4 M-tiles x 2 N-strips)
// Double-buffered async pipeline: copy chunk c+1 while WMMAs chew chunk c.
// ---------------------------------------------------------------------------
__global__ void __launch_bounds__(256) k_gather_gemm(
    const float* __restrict__ encodes, const float* __restrict__ Wc,
    const float* __restrict__ Wcb, const int* __restrict__ node_ids,
    float* __restrict__ h) {
  __shared__ float Xs[2][64][PITCH];
  __shared__ float Ws[2][64][PITCH];
  const int p    = blockIdx.x;
  const int nb   = blockIdx.y;          // 64 output columns per nb
  const int tid  = threadIdx.x;
  const int lane = tid & 31;
  const int wave = tid >> 5;
  const int wave_m = wave >> 1;         // 0..3
  const int wave_n = wave & 1;          // 0..1
  const int hi = lane >> 4, lo = lane & 15;
  const int rgrp = tid >> 3;            // staging: 32 rows per pass
  const int cgrp = (tid & 7) * 4;       // staging: float4 column within chunk

  // gather source row pointers (resolved once)
  const float* xsrc[2];
  const float* wsrc[2];
#pragma unroll
  for (int i = 0; i < 2; ++i) {
    const int r = rgrp + 32 * i;
    xsrc[i] = &encodes[(size_t)node_ids[p * BATCH + r] * E_DIM + cgrp];
    wsrc[i] = &Wc[(size_t)(nb * 64 + r) * E_DIM + cgrp];
  }

  auto stage = [&](int buf, int k0) {
#pragma unroll
    for (int i = 0; i < 2; ++i) {
      const int r = rgrp + 32 * i;
      async_copy16(xsrc[i] + k0, &Xs[buf][r][cgrp]);
      async_copy16(wsrc[i] + k0, &Ws[buf][r][cgrp]);
    }
  };

  v8f acc[2];
#pragma unroll
  for (int t = 0; t < 2; ++t) {
    const float bias = Wcb[nb * 64 + wave_n * 32 + t * 16 + lo];
#pragma unroll
    for (int v = 0; v < 8; ++v) acc[t][v] = bias;
  }

  stage(0, 0);
  for (int c = 0; c < NCHUNK; ++c) {
    wait_async0();                         // my copies for chunk c done
    __syncthreads();                       // everyone's copies visible
    if (c + 1 < NCHUNK) stage((c + 1) & 1, (c + 1) * KC);  // overlap
    gemm_chunk(Xs[c & 1], Ws[c & 1], wave_m, wave_n, lane, acc);
    __syncthreads();                       // all reads done before buf reuse
  }

#pragma unroll
  for (int t = 0; t < 2; ++t) {
    const int n = nb * 64 + wave_n * 32 + t * 16 + lo;
#pragma unroll
    for (int v = 0; v < 8; ++v) {
      const int m = wave_m * 16 + v + 8 * hi;
      h[((size_t)p * BATCH + m) * E_DIM + n] = acc[t][v];
    }
  }
}

// ---------------------------------------------------------------------------
// Phase 2: one tree level.  h[p] += (h[2p+1] + h[2p+2]) @ W_s^T + 2*b_s
// grid = (2^d, E_DIM/64); p = (2^d - 1) + blockIdx.x
// W tile via async-to-LDS; X tile (children sum) needs ALU -> ds_store path.
// ---------------------------------------------------------------------------
__global__ void __launch_bounds__(256) k_tree_level(
    const float* __restrict__ Wsum, const float* __restrict__ Wsb,
    float* __restrict__ h, int p0) {
  __shared__ float Xs[2][64][PITCH];
  __shared__ float Ws[2][64][PITCH];
  const int p    = p0 + blockIdx.x;
  const int nb   = blockIdx.y;
  const int tid  = threadIdx.x;
  const int lane = tid & 31;
  const int wave = tid >> 5;
  const int wave_m = wave >> 1;
  const int wave_n = wave & 1;
  const int hi = lane >> 4, lo = lane & 15;
  const int rgrp = tid >> 3;
  const int cgrp = (tid & 7) * 4;

  const float* hc0 = h + (size_t)(2 * p + 1) * BATCH * E_DIM;
  const float* hc1 = h + (size_t)(2 * p + 2) * BATCH * E_DIM;
  float*       hp  = h + (size_t)p * BATCH * E_DIM;

  auto stage_w = [&](int buf, int k0) {
#pragma unroll
    for (int i = 0; i < 2; ++i) {
      const int r = rgrp + 32 * i;
      async_copy16(&Wsum[(size_t)(nb * 64 + r) * E_DIM + k0 + cgrp],
                   &Ws[buf][r][cgrp]);
    }
  };
  auto stage_x = [&](int buf, int k0) {
#pragma unroll
    for (int i = 0; i < 2; ++i) {
      const int r = rgrp + 32 * i;
      const float4 a = *(const float4*)&hc0[(size_t)r * E_DIM + k0 + cgrp];
      const float4 b = *(const float4*)&hc1[(size_t)r * E_DIM + k0 + cgrp];
      Xs[buf][r][cgrp + 0] = a.x + b.x;
      Xs[buf][r][cgrp + 1] = a.y + b.y;
      Xs[buf][r][cgrp + 2] = a.z + b.z;
      Xs[buf][r][cgrp + 3] = a.w + b.w;
    }
  };

  v8f acc[2];
#pragma unroll
  for (int t = 0; t < 2; ++t) {
    const int n = nb * 64 + wave_n * 32 + t * 16 + lo;
    const float bias2 = 2.0f * Wsb[n];
#pragma unroll
    for (int v = 0; v < 8; ++v) {
      const int m = wave_m * 16 + v + 8 * hi;
      acc[t][v] = hp[(size_t)m * E_DIM + n] + bias2;
    }
  }

  stage_w(0, 0);
  for (int c = 0; c < NCHUNK; ++c) {
    stage_x(c & 1, c * KC);                // overlaps with in-flight async W
    wait_async0();                         // W chunk c arrived
    __syncthreads();                       // + X stores visible (dscnt waited)
    if (c + 1 < NCHUNK) stage_w((c + 1) & 1, (c + 1) * KC);
    gemm_chunk(Xs[c & 1], Ws[c & 1], wave_m, wave_n, lane, acc);
    __syncthreads();
  }

#pragma unroll
  for (int t = 0; t < 2; ++t) {
    const int n = nb * 64 + wave_n * 32 + t * 16 + lo;
#pragma unroll
    for (int v = 0; v < 8; ++v) {
      const int m = wave_m * 16 + v + 8 * hi;
      hp[(size_t)m * E_DIM + n] = acc[t][v];
    }
  }
}

// ---------------------------------------------------------------------------
// Phase 3a: out[rank(p)] = relu(h[p]) ; rank computed from heap-path bits.
// ---------------------------------------------------------------------------
__global__ void __launch_bounds__(256) k_postorder(const float* __restrict__ h,
                                                   float* __restrict__ out) {
  const int p = blockIdx.x;
  const int q = p + 1;
  const int depth = 31 - __clz(q);
  int S = P_NODES, r = 0;
  for (int i = depth - 1; i >= 0; --i) {
    S = (S - 1) >> 1;                   // child-subtree size
    if ((q >> i) & 1) r += S;           // right child: skip left subtree
  }
  r += S - 1;                           // subtree root is last in its postorder

  const float4* src = (const float4*)(h + (size_t)p * BATCH * E_DIM);
  float4*       dst = (float4*)(out + (size_t)r * BATCH * E_DIM);
  for (int i = threadIdx.x; i < BATCH * E_DIM / 4; i += 256) {
    float4 v = src[i];
    v.x = fmaxf(v.x, 0.0f);
    v.y = fmaxf(v.y, 0.0f);
    v.z = fmaxf(v.z, 0.0f);
    v.w = fmaxf(v.w, 0.0f);
    dst[i] = v;
  }
}

// ---------------------------------------------------------------------------
// Phase 3b: out_max[b,e] = max_p relu(h[p,b,e]) = max(0, max_p h[p,b,e])
// ---------------------------------------------------------------------------
__global__ void __launch_bounds__(256) k_maxred(const float* __restrict__ h,
                                                float* __restrict__ outmax) {
  const int i = blockIdx.x * 256 + threadIdx.x;   // 0 .. B*E-1
  const float* ptr = h + i;
  float m = 0.0f;                                  // relu floor
#pragma unroll 8
  for (int p = 0; p < P_NODES; ++p)
    m = fmaxf(m, ptr[(size_t)p * BATCH * E_DIM]);
  outmax[i] = m;
}

// ---------------------------------------------------------------------------
extern "C" void kernel_launch(void* const* d_in, const int* in_sizes, int n_in,
                              void* d_out, int out_size, void* d_ws,
                              size_t ws_size, hipStream_t stream) {
  (void)in_sizes; (void)n_in; (void)out_size; (void)ws_size;
  const float* encodes  = (const float*)d_in[0];
  const float* Wc       = (const float*)d_in[1];
  const float* Wcb      = (const float*)d_in[2];
  const float* Wsum     = (const float*)d_in[3];
  const float* Wsb      = (const float*)d_in[4];
  const int*   node_ids = (const int*)d_in[5];
  float* out = (float*)d_out;
  float* h   = (float*)d_ws;            // P*B*E f32 scratch (~64 MB)

  k_gather_gemm<<<dim3(P_NODES, E_DIM / 64), 256, 0, stream>>>(
      encodes, Wc, Wcb, node_ids, h);

  for (int d = TREE_DEPTH - 2; d >= 0; --d) {
    const int n = 1 << d;
    k_tree_level<<<dim3(n, E_DIM / 64), 256, 0, stream>>>(Wsum, Wsb, h, n - 1);
  }

  k_postorder<<<P_NODES, 256, 0, stream>>>(h, out);
  k_maxred<<<BATCH * E_DIM / 256, 256, 0, stream>>>(
      h, out + (size_t)P_NODES * BATCH * E_DIM);
}